// OnnxAttCovLoc_17489106829847
// MI455X (gfx1250) — compile-verified
//
#include <hip/hip_runtime.h>
#include <hip/hip_bf16.h>

#define H_LEN      16384
#define N_PREV     128
#define ATT_DIM    512
#define EPROJS     512
#define DUNITS     1024
#define NCH        10
#define KSIZE      201   // 2*100+1
#define KHALF      100
#define SCALING    2.0f

typedef __attribute__((ext_vector_type(2))) float v2f;
typedef __attribute__((ext_vector_type(8))) float v8f;

// ---------------------------------------------------------------------------
// K1: cov[h] = sum_n att_prev[n,h]; also copy att_prev rows into the output.
// float4-vectorized: 4096 threads, b128 loads/stores, stride-1 in h.
// ---------------------------------------------------------------------------
__global__ void k_cov_copy(const float4* __restrict__ att_prev,
                           float4* __restrict__ out_att,   // d_out + 512
                           float4* __restrict__ cov) {
  int h4 = blockIdx.x * blockDim.x + threadIdx.x;    // index in float4 units
  float4 s = make_float4(0.f, 0.f, 0.f, 0.f);
#pragma unroll 4
  for (int n = 0; n < N_PREV; ++n) {
    float4 v = att_prev[n * (H_LEN / 4) + h4];
    s.x += v.x; s.y += v.y; s.z += v.z; s.w += v.w;
    out_att[n * (H_LEN / 4) + h4] = v;
  }
  cov[h4] = s;
}

// ---------------------------------------------------------------------------
// K2: dproj[a] = sum_d dec_z[d] * mlp_dec_w[a, d].  One wave32 per row a.
// ---------------------------------------------------------------------------
__global__ void k_dec_proj(const float* __restrict__ dec_z,
                           const float* __restrict__ mlp_dec_w,
                           float* __restrict__ dproj) {
  int gtid = blockIdx.x * blockDim.x + threadIdx.x;
  int a    = gtid >> 5;
  int lane = threadIdx.x & 31;
  float s = 0.f;
  for (int d = lane; d < DUNITS; d += 32)
    s += dec_z[d] * mlp_dec_w[a * DUNITS + d];
  for (int off = 16; off > 0; off >>= 1)
    s += __shfl_down(s, off, 32);
  if (lane == 0) dproj[a] = s;
}

// ---------------------------------------------------------------------------
// K3: fused conv(10ch,k=201) + mlp_att + pre + dproj + tanh + gvec dot.
// One wave32 per h; 8 waves per block. Streams pre_compute_enc_h (32 MB)
// with float4 (b128) loads: each lane owns 4 consecutive att-dims.
// ---------------------------------------------------------------------------
__global__ void k_score(const float* __restrict__ cov,
                        const float* __restrict__ pre,
                        const float* __restrict__ loc_conv_w,   // [10][201]
                        const float* __restrict__ mlp_att_w,    // [512][10]
                        const float* __restrict__ dproj,        // [512]
                        const float* __restrict__ gvec_w,       // [512]
                        const float* __restrict__ gvec_b,       // [1]
                        const float* __restrict__ mask,         // [16384]
                        float* __restrict__ e) {                // [16384]
  __shared__ float s_lw[NCH * KSIZE];     // 2010 f
  __shared__ float s_aw[ATT_DIM * NCH];   // 5120 f
  __shared__ float s_dp[ATT_DIM];
  __shared__ float s_gv[ATT_DIM];

  for (int i = threadIdx.x; i < NCH * KSIZE; i += blockDim.x) s_lw[i] = loc_conv_w[i];
  for (int i = threadIdx.x; i < ATT_DIM * NCH; i += blockDim.x) s_aw[i] = mlp_att_w[i];
  for (int i = threadIdx.x; i < ATT_DIM; i += blockDim.x) {
    s_dp[i] = dproj[i];
    s_gv[i] = gvec_w[i];
  }
  __syncthreads();

  int wave = threadIdx.x >> 5;
  int lane = threadIdx.x & 31;
  int h = blockIdx.x * 8 + wave;

  // -- 201-tap, 10-channel conv at position h (lanes split the taps) --
  float ca[NCH];
#pragma unroll
  for (int c = 0; c < NCH; ++c) ca[c] = 0.f;
  for (int j = lane; j < KSIZE; j += 32) {
    int idx = h + j - KHALF;
    float cv = (idx >= 0 && idx < H_LEN) ? cov[idx] : 0.f;
#pragma unroll
    for (int c = 0; c < NCH; ++c) ca[c] += cv * s_lw[c * KSIZE + j];
  }
  float cc[NCH];
#pragma unroll
  for (int c = 0; c < NCH; ++c) {
    float v = ca[c];
    for (int off = 16; off > 0; off >>= 1) v += __shfl_down(v, off, 32);
    cc[c] = __shfl(v, 0, 32);               // broadcast reduced value
  }

  // -- 512-dim tanh + gvec dot; lane owns 4 consecutive dims per step --
  float acc = 0.f;
  for (int a0 = lane * 4; a0 < ATT_DIM; a0 += 128) {
    float4 p = *(const float4*)&pre[(size_t)h * ATT_DIM + a0];
    float pv[4] = {p.x, p.y, p.z, p.w};
#pragma unroll
    for (int q = 0; q < 4; ++q) {
      int a = a0 + q;
      float s = pv[q] + s_dp[a];
#pragma unroll
      for (int c = 0; c < NCH; ++c) s += cc[c] * s_aw[a * NCH + c];
      acc += s_gv[a] * tanhf(s);
    }
  }
  for (int off = 16; off > 0; off >>= 1) acc += __shfl_down(acc, off, 32);
  if (lane == 0) e[h] = acc + gvec_b[0] + mask[h];
}

// ---------------------------------------------------------------------------
// K4: single-block softmax over 16384 scores (in place: e -> w) and zero the
// context accumulator region for K5's atomics.
// ---------------------------------------------------------------------------
__global__ void k_softmax(float* __restrict__ ew,      // in: e, out: w (row 128)
                          float* __restrict__ c_out) { // d_out[0..511] -> 0
  __shared__ float s_red[32];
  int tid = threadIdx.x;
  float vals[16];
  float m = -3.402823466e38f;
#pragma unroll
  for (int i = 0; i < 16; ++i) {
    float v = SCALING * ew[tid + i * 1024];
    vals[i] = v;
    m = fmaxf(m, v);
  }
  for (int off = 16; off > 0; off >>= 1) m = fmaxf(m, __shfl_down(m, off, 32));
  if ((tid & 31) == 0) s_red[tid >> 5] = m;
  __syncthreads();
  if (tid < 32) {
    float v = s_red[tid];
    for (int off = 16; off > 0; off >>= 1) v = fmaxf(v, __shfl_down(v, off, 32));
    if (tid == 0) s_red[0] = v;
  }
  __syncthreads();
  m = s_red[0];
  __syncthreads();    // s_red reused below

  float sum = 0.f;
#pragma unroll
  for (int i = 0; i < 16; ++i) {
    float ev = __expf(vals[i] - m);
    vals[i] = ev;
    sum += ev;
  }
  for (int off = 16; off > 0; off >>= 1) sum += __shfl_down(sum, off, 32);
  if ((tid & 31) == 0) s_red[tid >> 5] = sum;
  __syncthreads();
  if (tid < 32) {
    float v = s_red[tid];
    for (int off = 16; off > 0; off >>= 1) v += __shfl_down(v, off, 32);
    if (tid == 0) s_red[0] = v;
  }
  __syncthreads();
  float inv = 1.f / s_red[0];
#pragma unroll
  for (int i = 0; i < 16; ++i) ew[tid + i * 1024] = vals[i] * inv;

  if (tid < EPROJS) c_out[tid] = 0.f;
}

// ---------------------------------------------------------------------------
// K5: c[a] = sum_h w[h] * enc_h[h, a]  via V_WMMA_F32_16X16X4_F32.
// A (16x4): only row M=0 populated with w[k..k+3]; lanes {0,16} supply
//   VGPR0/VGPR1 = K{0,1} / K{2,3} per ISA 7.12.2.
// B (4x16): VGPR0 = rows K0 (lanes 0-15) / K2 (lanes 16-31); VGPR1 = K1/K3.
// D (16x16 f32): VGPR0, lanes 0-15 == row M=0 == partial c for 16 columns.
// Branch-free A fragment: every lane loads w (1 KB L2-resident chunk) and
// non-row-0 lanes zero it with a multiply — no exec-mask save/restore in the
// inner loop. Bound by the 32 MB stream of enc_h.
// ---------------------------------------------------------------------------
#define KCHUNK 256

__global__ void k_ctx_wmma(const float* __restrict__ enc_h,
                           const float* __restrict__ w,
                           float* __restrict__ c_out) {
  const int lane  = threadIdx.x & 31;
  const int col   = blockIdx.x * 16 + (lane & 15);
  const int kbase = blockIdx.y * KCHUNK;
  const int hi    = (lane >> 4) & 1;       // lanes 16-31 handle K offsets 2,3
  const float wsel = ((lane & 15) == 0) ? 1.f : 0.f;   // only M=0 row of A

  v8f acc = {};
  for (int k = kbase; k < kbase + KCHUNK; k += 4) {
    const int kr = k + 2 * hi;
    v2f a;
    a.x = wsel * w[kr];          // unconditional load + select-by-multiply
    a.y = wsel * w[kr + 1];
    v2f b;
    b.x = enc_h[(size_t)kr * EPROJS + col];
    b.y = enc_h[(size_t)(kr + 1) * EPROJS + col];
    acc = __builtin_amdgcn_wmma_f32_16x16x4_f32(
        /*neg_a=*/false, a, /*neg_b=*/false, b,
        /*c_mod=*/(short)0, acc, /*reuse_a=*/false, /*reuse_b=*/false);
  }
  if (lane < 16) atomicAdd(&c_out[col], acc[0]);
}

// ---------------------------------------------------------------------------
extern "C" void kernel_launch(void* const* d_in, const int* in_sizes, int n_in,
                              void* d_out, int out_size, void* d_ws, size_t ws_size,
                              hipStream_t stream) {
  (void)in_sizes; (void)n_in; (void)out_size; (void)ws_size;

  const float* dec_z      = (const float*)d_in[0];
  const float* att_prev   = (const float*)d_in[1];
  const float* pre        = (const float*)d_in[2];
  const float* enc_h      = (const float*)d_in[3];
  const float* mask       = (const float*)d_in[4];
  const float* loc_conv_w = (const float*)d_in[5];
  const float* mlp_att_w  = (const float*)d_in[6];
  const float* mlp_dec_w  = (const float*)d_in[7];
  const float* gvec_w     = (const float*)d_in[8];
  const float* gvec_b     = (const float*)d_in[9];

  float* out   = (float*)d_out;
  float* c_out = out;                    // [512]
  float* out_att = out + EPROJS;         // [129][16384]
  float* ew_row  = out_att + (size_t)N_PREV * H_LEN;  // row 128: e then w

  float* ws    = (float*)d_ws;
  float* cov   = ws;                     // [16384]
  float* dproj = ws + H_LEN;             // [512]

  // K1: coverage sum + copy old attention rows (float4 path)
  k_cov_copy<<<(H_LEN / 4) / 256, 256, 0, stream>>>(
      (const float4*)att_prev, (float4*)out_att, (float4*)cov);
  // K2: decoder projection (512 wave32s)
  k_dec_proj<<<(ATT_DIM * 32) / 256, 256, 0, stream>>>(dec_z, mlp_dec_w, dproj);
  // K3: fused conv+mlp_att+tanh+gvec -> e (one wave per h)
  k_score<<<H_LEN / 8, 256, 0, stream>>>(cov, pre, loc_conv_w, mlp_att_w,
                                         dproj, gvec_w, gvec_b, mask, ew_row);
  // K4: softmax in place (e -> w) + zero c accumulators
  k_softmax<<<1, 1024, 0, stream>>>(ew_row, c_out);
  // K5: context vector via f32 WMMA, atomic partial sums
  k_ctx_wmma<<<dim3(EPROJS / 16, H_LEN / KCHUNK), 32, 0, stream>>>(enc_h, ew_row, c_out);
}